// GINLayer_80161269613390
// MI455X (gfx1250) — compile-verified
//
#include <hip/hip_runtime.h>

#define DF 128
#define PADF 4
#define LSTRIDE (DF + PADF)

typedef __attribute__((ext_vector_type(2))) float v2f;
typedef __attribute__((ext_vector_type(8))) float v8f;

// order-preserving f32 -> u32 key (monotone under unsigned compare)
__device__ __forceinline__ unsigned fenc(float f) {
    unsigned u = __float_as_uint(f);
    return u ^ ((unsigned)((int)u >> 31) | 0x80000000u);
}
// key 0 is the "no edge" sentinel (below key(-inf) = 0x007FFFFF)
__device__ __forceinline__ float fdec(unsigned k) {
    if (k == 0u) return 0.0f;
    unsigned u = (k & 0x80000000u) ? (k ^ 0x80000000u) : ~k;
    return __uint_as_float(u);
}

__global__ __launch_bounds__(256) void k_init_keys(unsigned* __restrict__ keys, int n) {
    int i = blockIdx.x * 256 + threadIdx.x;
    if (i < n) keys[i] = 0u;
}

// one thread per (edge, feature): gather feats[src[e]][f], atomic-max into dst row
__global__ __launch_bounds__(256) void k_edge_max(const float* __restrict__ feats,
                                                  const int* __restrict__ src,
                                                  const int* __restrict__ dst,
                                                  unsigned* __restrict__ keys,
                                                  long long total) {
    long long i = (long long)blockIdx.x * 256 + threadIdx.x;
    if (i >= total) return;
    int e = (int)(i >> 7);
    int f = (int)(i & (DF - 1));
    float v = feats[(long long)src[e] * DF + f];
    atomicMax(&keys[(long long)dst[e] * DF + f], fenc(v));
}

// Block = 256 threads = 8 waves; block tile = 16 rows x 128 cols.
// Wave w computes the 16x16 tile at columns [16w, 16w+16) via V_WMMA_F32_16X16X4_F32.
// MODE 0: A = feats + decode(aggKeys)   (GIN input fuse)
// MODE 1: A = relu(bnA*Z + bnC)         (BN+ReLU of previous layer fused into load)
// Epilogue: +bias, store Z, emit per-block column sum/sumsq partials (deterministic).
template <int MODE>
__global__ __launch_bounds__(256) void k_gemm(const float* __restrict__ A,
                                              const unsigned* __restrict__ aggKeys,
                                              const float* __restrict__ bnA,
                                              const float* __restrict__ bnC,
                                              const float* __restrict__ W,
                                              const float* __restrict__ bias,
                                              float* __restrict__ Z,
                                              float* __restrict__ pSum,
                                              float* __restrict__ pSq) {
    __shared__ float As[16 * LSTRIDE];
    const int tid = threadIdx.x;
    const int rowBase = blockIdx.x * 16;

    // coalesced fill of the 16x128 A tile, with fused elementwise pre-transform
#pragma unroll
    for (int i = 0; i < 8; ++i) {
        int off = i * 256 + tid;
        int r = off >> 7;
        int ck = off & (DF - 1);
        long long gidx = (long long)(rowBase + r) * DF + ck;
        float v = A[gidx];
        if (MODE == 0) v += fdec(aggKeys[gidx]);
        else           v = fmaxf(fmaf(bnA[ck], v, bnC[ck]), 0.0f);
        As[r * LSTRIDE + ck] = v;
    }
    __syncthreads();

    const int wave = tid >> 5;
    const int lane = tid & 31;
    const int half = lane >> 4;   // K sub-group per ISA A/B 16x4 layout
    const int l16  = lane & 15;
    const int col  = wave * 16 + l16;

    v8f acc = {0.f, 0.f, 0.f, 0.f, 0.f, 0.f, 0.f, 0.f};
#pragma unroll 8
    for (int kb = 0; kb < DF; kb += 4) {
        int kk = kb + half * 2;
        v2f a, b;
        // A 16x4 fp32 frag: lanes 0-15 hold K=kb..kb+1, lanes 16-31 hold K=kb+2..kb+3
        a.x = As[l16 * LSTRIDE + kk];
        a.y = As[l16 * LSTRIDE + kk + 1];
        // B 4x16 fp32 frag: vgpr v, lane-half h -> K = kb + 2h + v, N = l16
        b.x = W[kk * DF + col];
        b.y = W[(kk + 1) * DF + col];
        acc = __builtin_amdgcn_wmma_f32_16x16x4_f32(false, a, false, b,
                                                    (short)0, acc, false, false);
    }

    // epilogue: bias add, store, column stats (C/D layout: vgpr i, half h -> M = i + 8h)
    float bcol = bias[col];
    float s = 0.f, q = 0.f;
#pragma unroll
    for (int i = 0; i < 8; ++i) {
        float z = acc[i] + bcol;
        int m = i + half * 8;
        Z[(long long)(rowBase + m) * DF + col] = z;
        s += z;
        q += z * z;
    }
    s += __shfl_xor(s, 16, 32);
    q += __shfl_xor(q, 16, 32);
    if (lane < 16) {
        pSum[(long long)blockIdx.x * DF + col] = s;
        pSq [(long long)blockIdx.x * DF + col] = q;
    }
}

// fixed-order partial reduction -> BN scale/shift coeffs (deterministic)
__global__ __launch_bounds__(128) void k_reduce_bn(const float* __restrict__ pSum,
                                                   const float* __restrict__ pSq,
                                                   int nBlocks, float invN,
                                                   const float* __restrict__ gamma,
                                                   const float* __restrict__ beta,
                                                   float* __restrict__ bnA,
                                                   float* __restrict__ bnC) {
    int c = threadIdx.x;
    float s = 0.f, q = 0.f;
    for (int b = 0; b < nBlocks; ++b) {
        s += pSum[(long long)b * DF + c];
        q += pSq [(long long)b * DF + c];
    }
    float m = s * invN;
    float v = fmaf(-m, m, q * invN);
    float a = gamma[c] * rsqrtf(v + 1e-5f);
    bnA[c] = a;
    bnC[c] = fmaf(-m, a, beta[c]);
}

// in-place relu(BN2) + partial stats for BN3; block tile 16 rows x 128 cols
__global__ __launch_bounds__(256) void k_bnrelu_stats(float* __restrict__ Z,
                                                      const float* __restrict__ bnA,
                                                      const float* __restrict__ bnC,
                                                      float* __restrict__ pSum,
                                                      float* __restrict__ pSq) {
    __shared__ float sh[256];
    __shared__ float shq[256];
    int tid = threadIdx.x;
    int c = tid & (DF - 1);
    int rowBase = blockIdx.x * 16 + (tid >> 7) * 8;
    float a = bnA[c], cc = bnC[c];
    float s = 0.f, q = 0.f;
#pragma unroll
    for (int r = 0; r < 8; ++r) {
        long long idx = (long long)(rowBase + r) * DF + c;
        float h = fmaxf(fmaf(a, Z[idx], cc), 0.0f);
        Z[idx] = h;
        s += h;
        q += h * h;
    }
    sh[tid] = s;
    shq[tid] = q;
    __syncthreads();
    if (tid < 128) {
        pSum[(long long)blockIdx.x * DF + tid] = sh[tid] + sh[tid + 128];
        pSq [(long long)blockIdx.x * DF + tid] = shq[tid] + shq[tid + 128];
    }
}

__global__ __launch_bounds__(256) void k_final(const float* __restrict__ H,
                                               const float* __restrict__ bnA,
                                               const float* __restrict__ bnC,
                                               float* __restrict__ out, long long n) {
    long long i = (long long)blockIdx.x * 256 + threadIdx.x;
    if (i >= n) return;
    int c = (int)(i & (DF - 1));
    out[i] = fmaf(bnA[c], H[i], bnC[c]);
}

extern "C" void kernel_launch(void* const* d_in, const int* in_sizes, int n_in,
                              void* d_out, int out_size, void* d_ws, size_t ws_size,
                              hipStream_t stream) {
    (void)n_in; (void)out_size; (void)ws_size;
    const float* feats = (const float*)d_in[0];
    const int*   src   = (const int*)d_in[1];
    const int*   dst   = (const int*)d_in[2];
    const float* W1    = (const float*)d_in[3];
    const float* b1    = (const float*)d_in[4];
    const float* g1    = (const float*)d_in[5];
    const float* be1   = (const float*)d_in[6];
    const float* W2    = (const float*)d_in[7];
    const float* b2    = (const float*)d_in[8];
    const float* g2    = (const float*)d_in[9];
    const float* be2   = (const float*)d_in[10];
    const float* g3    = (const float*)d_in[11];
    const float* be3   = (const float*)d_in[12];
    float* out = (float*)d_out;

    const int nNodes = in_sizes[0] / DF;           // 50000
    const int nEdges = in_sizes[1];                // 800000
    const int nBlk   = nNodes / 16;                // 3125 (50000 % 16 == 0)
    const long long nElems = (long long)nNodes * DF;

    char* ws = (char*)d_ws;
    unsigned* aggKeys = (unsigned*)ws;                          // 25.6 MB
    float* Z1   = (float*)(ws + (size_t)nElems * 4);            // 25.6 MB
    float* Z2   = (float*)(ws + (size_t)nElems * 8);            // 25.6 MB
    float* pSum = (float*)(ws + (size_t)nElems * 12);           // nBlk*128 f32
    float* pSq  = pSum + (size_t)nBlk * DF;                     // nBlk*128 f32
    float* cf   = pSq  + (size_t)nBlk * DF;                     // 6*128 f32
    float *a1 = cf,        *c1 = cf + DF;
    float *a2 = cf + 2*DF, *c2 = cf + 3*DF;
    float *a3 = cf + 4*DF, *c3 = cf + 5*DF;

    const float invN = 1.0f / (float)nNodes;

    k_init_keys<<<(unsigned)((nElems + 255) / 256), 256, 0, stream>>>(aggKeys, (int)nElems);

    const long long eTot = (long long)nEdges * DF;
    k_edge_max<<<(unsigned)((eTot + 255) / 256), 256, 0, stream>>>(feats, src, dst, aggKeys, eTot);

    k_gemm<0><<<nBlk, 256, 0, stream>>>(feats, aggKeys, nullptr, nullptr, W1, b1, Z1, pSum, pSq);
    k_reduce_bn<<<1, 128, 0, stream>>>(pSum, pSq, nBlk, invN, g1, be1, a1, c1);

    k_gemm<1><<<nBlk, 256, 0, stream>>>(Z1, nullptr, a1, c1, W2, b2, Z2, pSum, pSq);
    k_reduce_bn<<<1, 128, 0, stream>>>(pSum, pSq, nBlk, invN, g2, be2, a2, c2);

    k_bnrelu_stats<<<nBlk, 256, 0, stream>>>(Z2, a2, c2, pSum, pSq);
    k_reduce_bn<<<1, 128, 0, stream>>>(pSum, pSq, nBlk, invN, g3, be3, a3, c3);

    k_final<<<(unsigned)((nElems + 255) / 256), 256, 0, stream>>>(Z2, a3, c3, out, nElems);
}